// PASGeN_21131239096640
// MI455X (gfx1250) — compile-verified
//
#include <hip/hip_runtime.h>
#include <hip/hip_bf16.h>

typedef float v2f __attribute__((ext_vector_type(2)));
typedef float v8f __attribute__((ext_vector_type(8)));

#define D_EMB   128
#define D_STATE 64
#define D_EDGE  32
#define D_H     192   // D_EMB + D_STATE
#define H1      32    // hidden width of fc_attn

// ---- order-preserving float <-> uint mapping for atomicMax on floats ----
__device__ __forceinline__ unsigned fenc(float f) {
    unsigned u = __float_as_uint(f);
    return (u & 0x80000000u) ? ~u : (u | 0x80000000u);
}
__device__ __forceinline__ float fdec(unsigned u) {
    unsigned b = (u & 0x80000000u) ? (u & 0x7FFFFFFFu) : ~u;
    return __uint_as_float(b);
}

__device__ __forceinline__ int lower_bound_i(const int* __restrict__ arr, int n, int key) {
    int lo = 0, hi = n;
    while (lo < hi) {
        int mid = (lo + hi) >> 1;
        if (arr[mid] < key) lo = mid + 1; else hi = mid;
    }
    return lo;
}

// ---------------------------------------------------------------------------
// K0: zero the per-graph reduction buffers (ws is poisoned to 0xAA).
// ---------------------------------------------------------------------------
__global__ void init_seg_kernel(unsigned* __restrict__ seg_max_u,
                                float* __restrict__ seg_sum, int B) {
    int i = blockIdx.x * blockDim.x + threadIdx.x;
    if (i < B) { seg_max_u[i] = 0u; seg_sum[i] = 0.0f; }
}

// ---------------------------------------------------------------------------
// K1: attention logits via WMMA.
// One wave32 per 16-node tile: [16 x 192] @ [192 x 32] with
// V_WMMA_F32_16X16X4_F32 (two 16-col accumulators, 48 K-steps of 4),
// then leaky_relu, dot with W2 via a 16-lane butterfly reduce,
// fused atomicMax into per-graph segment max.
// ---------------------------------------------------------------------------
__global__ __launch_bounds__(256) void attn_logits_kernel(
    const float* __restrict__ z_atom,     // [N,128]
    const float* __restrict__ state_emb,  // [10,64]
    const int*   __restrict__ state,      // [B]
    const int*   __restrict__ batch,      // [N] sorted
    const float* __restrict__ W1,         // [192,32]
    const float* __restrict__ b1,         // [32]
    const float* __restrict__ W2,         // [32,1]
    const float* __restrict__ b2,         // [1]
    float*       __restrict__ logits,     // [N] out
    unsigned*    __restrict__ seg_max_u,  // [B] atomicMax target
    int N)
{
    const int gwave = (blockIdx.x * blockDim.x + threadIdx.x) >> 5;  // tile id
    const int lane  = threadIdx.x & 31;
    const int tile  = gwave;
    if (tile * 16 >= N) return;            // wave-uniform: EXEC stays all-ones

    const int m    = lane & 15;            // row within tile (A), col (B/C)
    const int half = lane >> 4;            // K-pair selector per ISA layout

    int node = tile * 16 + m;
    if (node >= N) node = N - 1;           // safe clamp (N % 16 == 0 normally)

    const float* zrow = z_atom + (size_t)node * D_EMB;
    const int    b_of = batch[node];
    const float* srow = state_emb + (size_t)state[b_of] * D_STATE;

    v8f acc0 = {0.f,0.f,0.f,0.f,0.f,0.f,0.f,0.f};
    v8f acc1 = {0.f,0.f,0.f,0.f,0.f,0.f,0.f,0.f};

    // A 16x4 f32 layout: VGPR0 = K {0 | 2}, VGPR1 = K {1 | 3} across half-waves.
    #pragma unroll
    for (int k0 = 0; k0 < D_H; k0 += 4) {
        const int ka = k0 + 2 * half;                       // ka, ka+1 same side of 128
        const float* hsrc = (ka < D_EMB) ? (zrow + ka) : (srow + (ka - D_EMB));
        v2f A;  A.x = hsrc[0];                A.y = hsrc[1];
        v2f B0; B0.x = W1[ka * H1 + m];       B0.y = W1[(ka + 1) * H1 + m];
        v2f B1; B1.x = W1[ka * H1 + 16 + m];  B1.y = W1[(ka + 1) * H1 + 16 + m];
        acc0 = __builtin_amdgcn_wmma_f32_16x16x4_f32(false, A, false, B0,
                                                     (short)0, acc0, false, false);
        acc1 = __builtin_amdgcn_wmma_f32_16x16x4_f32(false, A, false, B1,
                                                     (short)0, acc1, false, false);
    }

    const float w2a = W2[m],      w2b = W2[16 + m];
    const float b1a = b1[m],      b1b = b1[16 + m];
    const float bias2 = b2[0];

    // C/D layout: VGPR r, lanes 0-15 -> M=r, lanes 16-31 -> M=r+8; N = m.
    #pragma unroll
    for (int r = 0; r < 8; ++r) {
        float a0 = acc0[r] + b1a;  a0 = (a0 > 0.f) ? a0 : 0.01f * a0;  // leaky_relu
        float a1 = acc1[r] + b1b;  a1 = (a1 > 0.f) ? a1 : 0.01f * a1;
        float t = a0 * w2a + a1 * w2b;
        t += __shfl_xor(t, 1, 32);
        t += __shfl_xor(t, 2, 32);
        t += __shfl_xor(t, 4, 32);
        t += __shfl_xor(t, 8, 32);     // each 16-lane half now holds its row sum
        if (m == 0) {
            int nd = tile * 16 + r + 8 * half;
            if (nd < N) {
                float lg = t + bias2;
                logits[nd] = lg;
                atomicMax(&seg_max_u[batch[nd]], fenc(lg));
            }
        }
    }
}

// ---------------------------------------------------------------------------
// K2: ex = exp(logit - seg_max[batch]); seg_sum += ex  (ex overwrites logits)
// ---------------------------------------------------------------------------
__global__ void softmax_exp_kernel(const int* __restrict__ batch,
                                   const unsigned* __restrict__ seg_max_u,
                                   float* __restrict__ logits_ex,
                                   float* __restrict__ seg_sum, int N)
{
    int n = blockIdx.x * blockDim.x + threadIdx.x;
    if (n >= N) return;
    int b = batch[n];
    float ex = expf(logits_ex[n] - fdec(seg_max_u[b]));
    logits_ex[n] = ex;
    atomicAdd(&seg_sum[b], ex);
}

// ---------------------------------------------------------------------------
// K3: one block per graph. x[b] = [ z_mol[b] | sum_n attn(n) * z_atom[n] ].
// batch sorted -> binary-search the segment; coalesced column reads.
// ---------------------------------------------------------------------------
__global__ __launch_bounds__(128) void pool_kernel(
    const float* __restrict__ z_mol,     // [B,128]
    const float* __restrict__ z_atom,    // [N,128]
    const int*   __restrict__ batch,     // [N] sorted
    const float* __restrict__ ex,        // [N]
    const float* __restrict__ seg_sum,   // [B]
    float*       __restrict__ x,         // [B,256] out
    int N)
{
    const int b = blockIdx.x;
    const int d = threadIdx.x;           // 0..127
    const int start = lower_bound_i(batch, N, b);
    const int end   = lower_bound_i(batch, N, b + 1);
    const float inv = 1.0f / (seg_sum[b] + 1e-16f);
    float acc = 0.0f;
    for (int n = start; n < end; ++n)
        acc += ex[n] * z_atom[(size_t)n * D_EMB + d];
    x[(size_t)b * 256 + d]       = z_mol[(size_t)b * D_EMB + d];
    x[(size_t)b * 256 + 128 + d] = acc * inv;
}

// ---------------------------------------------------------------------------
// K4: edge_attr_out[e] = [ edge_attr[e] (32) | state_emb[state[batch[ei0[e]]]] (64) ]
// 24 float4-threads per edge, fully vectorized 128-bit traffic.
// ---------------------------------------------------------------------------
__global__ void edge_concat_kernel(const float* __restrict__ edge_attr, // [E,32]
                                   const float* __restrict__ state_emb, // [10,64]
                                   const int*   __restrict__ state,     // [B]
                                   const int*   __restrict__ ei0,       // [E]
                                   const int*   __restrict__ batch,     // [N]
                                   float*       __restrict__ out_edges, // [E,96]
                                   long long total)                     // E*24
{
    long long t = (long long)blockIdx.x * blockDim.x + threadIdx.x;
    if (t >= total) return;
    int e = (int)(t / 24);
    int q = (int)(t % 24);
    float4 v;
    if (q < 8) {
        v = ((const float4*)edge_attr)[(size_t)e * 8 + q];
    } else {
        int s = state[batch[ei0[e]]];
        v = ((const float4*)state_emb)[(size_t)s * 16 + (q - 8)];
    }
    ((float4*)out_edges)[(size_t)e * 24 + q] = v;
}

// ---------------------------------------------------------------------------
extern "C" void kernel_launch(void* const* d_in, const int* in_sizes, int n_in,
                              void* d_out, int out_size, void* d_ws, size_t ws_size,
                              hipStream_t stream) {
    const float* edge_attr = (const float*)d_in[0];
    const float* z_mol     = (const float*)d_in[1];
    const float* z_atom    = (const float*)d_in[2];
    const float* state_emb = (const float*)d_in[3];
    const float* W1        = (const float*)d_in[4];
    const float* b1        = (const float*)d_in[5];
    const float* W2        = (const float*)d_in[6];
    const float* b2        = (const float*)d_in[7];
    const int*   state     = (const int*)d_in[8];
    const int*   ei0       = (const int*)d_in[9];
    const int*   batch     = (const int*)d_in[10];

    const int E = in_sizes[0] / D_EDGE;
    const int B = in_sizes[1] / D_EMB;
    const int N = in_sizes[2] / D_EMB;

    float* x         = (float*)d_out;               // [B,256]
    float* out_edges = x + (size_t)B * 256;         // [E,96]

    float*    logits_ex = (float*)d_ws;             // [N] logits -> ex (in place)
    unsigned* seg_max_u = (unsigned*)(logits_ex + N);   // [B]
    float*    seg_sum   = (float*)(seg_max_u + B);      // [B]

    // K0: init segment reduction buffers
    init_seg_kernel<<<(B + 255) / 256, 256, 0, stream>>>(seg_max_u, seg_sum, B);

    // K1: WMMA attention logits + fused segment max
    const int tiles = (N + 15) / 16;                // one wave32 per tile
    attn_logits_kernel<<<(tiles + 7) / 8, 256, 0, stream>>>(
        z_atom, state_emb, state, batch, W1, b1, W2, b2,
        logits_ex, seg_max_u, N);

    // K2: exp + segment sum
    softmax_exp_kernel<<<(N + 255) / 256, 256, 0, stream>>>(
        batch, seg_max_u, logits_ex, seg_sum, N);

    // K3: weighted pool + concat with z_mol
    pool_kernel<<<B, 128, 0, stream>>>(z_mol, z_atom, batch,
                                       logits_ex, seg_sum, x, N);

    // K4: edge feature concat (independent of K1-K3, stream-ordered anyway)
    const long long tot = (long long)E * 24;
    edge_concat_kernel<<<(int)((tot + 255) / 256), 256, 0, stream>>>(
        edge_attr, state_emb, state, ei0, batch, out_edges, tot);
}